// GCN_EW_13400297963542
// MI455X (gfx1250) — compile-verified
//
#include <hip/hip_runtime.h>
#include <hip/hip_bf16.h>

#define NN 50000
#define NE 640000
#define HID 128
#define ODIM 16
#define BATCH 2
#define TOTAL_ROWS (BATCH * NN)
#define BN_EPS 1e-5f

typedef __attribute__((ext_vector_type(2))) float v2f;
typedef __attribute__((ext_vector_type(8))) float v8f;

// ---------------- graph normalization ----------------

__global__ void k_init_deg(float* __restrict__ deg) {
  int i = blockIdx.x * blockDim.x + threadIdx.x;
  if (i < NN) deg[i] = 1.0f;   // self-loop weight 1
}

__global__ void k_deg_accum(const int* __restrict__ ei, const float* __restrict__ p,
                            float* __restrict__ deg) {
  int e = blockIdx.x * blockDim.x + threadIdx.x;
  if (e < NE) atomicAdd(&deg[ei[NE + e]], __expf(p[e]));
}

__global__ void k_dinv(float* __restrict__ deg) {
  int i = blockIdx.x * blockDim.x + threadIdx.x;
  if (i < NN) deg[i] = rsqrtf(deg[i]);   // deg >= 1 always
}

__global__ void k_norm(const int* __restrict__ ei, const float* __restrict__ p,
                       const float* __restrict__ dinv, float* __restrict__ nrm) {
  int e = blockIdx.x * blockDim.x + threadIdx.x;
  if (e < NE) nrm[e] = dinv[ei[e]] * __expf(p[e]) * dinv[ei[NE + e]];
}

// ---------------- WMMA GEMM: [TOTAL_ROWS,128] x [128,128] ----------------
// Optional fused input transform a = a*scale[k] + shift[k] (BatchNorm of prior layer).
// Block = 256 threads = 8 wave32; each wave: 16 rows x 128 cols (8 WMMA tiles).

__launch_bounds__(256)
__global__ void k_gemm128(const float* __restrict__ A, const float* __restrict__ W,
                          float* __restrict__ out,
                          const float* __restrict__ scale, const float* __restrict__ shift,
                          int hasBN) {
  __shared__ float ldsW[128 * 128];   // 64 KB
  int tid = threadIdx.x;

  const float4* Wv = (const float4*)W;
  float4* Lv = (float4*)ldsW;
  for (int i = tid; i < 128 * 128 / 4; i += 256) Lv[i] = Wv[i];
  __syncthreads();

  int wave = tid >> 5;
  int lane = tid & 31;
  int half = lane >> 4;      // 0: K pair {0,1}; 1: K pair {2,3} of each 4-step
  int l16  = lane & 15;
  int c0   = half * 2;

  long rowBase = (long)blockIdx.x * 128 + wave * 16;
  long aRow = rowBase + l16;
  if (aRow >= TOTAL_ROWS) aRow = TOTAL_ROWS - 1;   // clamp; stores are guarded
  const float* Ar = A + aRow * 128;

  // Load all A fragments once (K = 128 -> 32 wmma steps), fusing BN transform.
  v2f afrag[32];
#pragma unroll
  for (int t = 0; t < 32; ++t) {
    int k = 4 * t + c0;
    float a0 = Ar[k];
    float a1 = Ar[k + 1];
    if (hasBN) {
      a0 = a0 * scale[k] + shift[k];
      a1 = a1 * scale[k + 1] + shift[k + 1];
    }
    afrag[t].x = a0;
    afrag[t].y = a1;
  }

  for (int ct = 0; ct < 8; ++ct) {
    int col = ct * 16 + l16;
    v8f c = {};
#pragma unroll
    for (int t = 0; t < 32; ++t) {
      int k = 4 * t + c0;
      v2f b;
      b.x = ldsW[k * 128 + col];
      b.y = ldsW[(k + 1) * 128 + col];
      c = __builtin_amdgcn_wmma_f32_16x16x4_f32(false, afrag[t], false, b,
                                                (short)0, c, false, false);
    }
#pragma unroll
    for (int v = 0; v < 8; ++v) {
      long r = rowBase + half * 8 + v;   // C layout: VGPR v -> M = v (+8 for upper half)
      if (r < TOTAL_ROWS) out[r * 128 + col] = c[v];
    }
  }
}

// ---------------- aggregation: self-loop + bias init, then edge scatter ----------------

__global__ void k_agg_init(const float* __restrict__ xw, const float* __restrict__ dinv,
                           const float* __restrict__ bias, float* __restrict__ agg) {
  long g = (long)blockIdx.x * blockDim.x + threadIdx.x;
  if (g < (long)TOTAL_ROWS * 128) {
    int f = (int)(g & 127);
    long row = g >> 7;
    int node = (int)(row % NN);
    float d = dinv[node];
    agg[g] = bias[f] + xw[g] * d * d;
  }
}

__global__ void k_scatter(const int* __restrict__ ei, const float* __restrict__ nrm,
                          const float* __restrict__ xw, float* __restrict__ agg) {
  long gwave = ((long)blockIdx.x * blockDim.x + threadIdx.x) >> 5;
  int lane = threadIdx.x & 31;
  if (gwave >= (long)BATCH * NE) return;
  int b = (int)(gwave / NE);
  int e = (int)(gwave - (long)b * NE);
  int r = ei[e];
  int c = ei[NE + e];
  float w = nrm[e];
  const float* src = xw + ((long)b * NN + r) * 128;
  float* dst = agg + ((long)b * NN + c) * 128;
#pragma unroll
  for (int j = 0; j < 4; ++j)
    atomicAdd(&dst[lane + 32 * j], src[lane + 32 * j] * w);
}

// ---------------- ReLU (in place) + BN statistics ----------------

__global__ void k_zero_stats(float* __restrict__ stats) {
  int i = threadIdx.x;
  if (i < 256) stats[i] = 0.0f;   // [0..127]=sum, [128..255]=sumsq
}

__global__ void k_relu_stats(float* __restrict__ buf, float* __restrict__ stats) {
  int f = threadIdx.x;   // blockDim = 128
  float s = 0.0f, sq = 0.0f;
  for (int r = blockIdx.x; r < TOTAL_ROWS; r += gridDim.x) {
    long idx = (long)r * 128 + f;
    float v = buf[idx];
    v = v > 0.0f ? v : 0.0f;
    buf[idx] = v;
    s += v;
    sq += v * v;
  }
  atomicAdd(&stats[f], s);
  atomicAdd(&stats[128 + f], sq);
}

__global__ void k_bn_finalize(const float* __restrict__ stats,
                              const float* __restrict__ gamma, const float* __restrict__ beta,
                              float* __restrict__ scale, float* __restrict__ shift) {
  int f = threadIdx.x;   // 128
  float invn = 1.0f / (float)TOTAL_ROWS;
  float mu = stats[f] * invn;
  float var = stats[128 + f] * invn - mu * mu;   // biased variance
  float s = gamma[f] * rsqrtf(var + BN_EPS);
  scale[f] = s;
  shift[f] = beta[f] - mu * s;
}

// ---------------- final WMMA GEMM: [TOTAL_ROWS,128] x [128,16] + bc ----------------

__launch_bounds__(256)
__global__ void k_gemm_out(const float* __restrict__ A, const float* __restrict__ Wc,
                           const float* __restrict__ bc,
                           const float* __restrict__ scale, const float* __restrict__ shift,
                           float* __restrict__ out) {
  __shared__ float ldsW[128 * 16];
  __shared__ float ldsB[16];
  int tid = threadIdx.x;
  for (int i = tid; i < 128 * 16; i += 256) ldsW[i] = Wc[i];
  if (tid < 16) ldsB[tid] = bc[tid];
  __syncthreads();

  int wave = tid >> 5, lane = tid & 31, half = lane >> 4, l16 = lane & 15;
  int c0 = half * 2;
  long rowBase = (long)blockIdx.x * 128 + wave * 16;
  long aRow = rowBase + l16;
  if (aRow >= TOTAL_ROWS) aRow = TOTAL_ROWS - 1;
  const float* Ar = A + aRow * 128;

  v8f c = {};
#pragma unroll
  for (int t = 0; t < 32; ++t) {
    int k = 4 * t + c0;
    v2f a, b;
    a.x = Ar[k] * scale[k] + shift[k];
    a.y = Ar[k + 1] * scale[k + 1] + shift[k + 1];
    b.x = ldsW[k * 16 + l16];
    b.y = ldsW[(k + 1) * 16 + l16];
    c = __builtin_amdgcn_wmma_f32_16x16x4_f32(false, a, false, b, (short)0, c, false, false);
  }
#pragma unroll
  for (int v = 0; v < 8; ++v) {
    long r = rowBase + half * 8 + v;
    if (r < TOTAL_ROWS) out[r * ODIM + l16] = c[v] + ldsB[l16];
  }
}

// ---------------- host launcher ----------------

static inline size_t alignup(size_t x) { return (x + 255) & ~(size_t)255; }

extern "C" void kernel_launch(void* const* d_in, const int* in_sizes, int n_in,
                              void* d_out, int out_size, void* d_ws, size_t ws_size,
                              hipStream_t stream) {
  const float* x   = (const float*)d_in[0];
  const int*   ei  = (const int*)d_in[1];
  const float* ewp = (const float*)d_in[2];
  const float* W1  = (const float*)d_in[3];
  const float* b1  = (const float*)d_in[4];
  const float* W2  = (const float*)d_in[5];
  const float* b2  = (const float*)d_in[6];
  const float* g1  = (const float*)d_in[7];
  const float* be1 = (const float*)d_in[8];
  const float* g2  = (const float*)d_in[9];
  const float* be2 = (const float*)d_in[10];
  const float* Wc  = (const float*)d_in[11];
  const float* bc  = (const float*)d_in[12];
  float* out = (float*)d_out;

  char* w = (char*)d_ws;
  float* deg   = (float*)w; w += alignup((size_t)NN * 4);          // becomes dinv in place
  float* nrm   = (float*)w; w += alignup((size_t)NE * 4);
  float* stats = (float*)w; w += alignup(256 * 4);
  float* scale = (float*)w; w += alignup(128 * 4);
  float* shift = (float*)w; w += alignup(128 * 4);
  float* bufA  = (float*)w; w += alignup((size_t)TOTAL_ROWS * 128 * 4);
  float* bufB  = (float*)w; w += alignup((size_t)TOTAL_ROWS * 128 * 4);

  const int gemmBlocks = (TOTAL_ROWS + 127) / 128;                 // 782
  const int ewBlocks   = ((long)BATCH * NE / 8);                   // 160000 (8 edges/block)
  const int initBlocks = (int)(((long)TOTAL_ROWS * 128 + 255) / 256);

  // graph normalization
  k_init_deg <<<(NN + 255) / 256, 256, 0, stream>>>(deg);
  k_deg_accum<<<(NE + 255) / 256, 256, 0, stream>>>(ei, ewp, deg);
  k_dinv     <<<(NN + 255) / 256, 256, 0, stream>>>(deg);
  k_norm     <<<(NE + 255) / 256, 256, 0, stream>>>(ei, ewp, deg, nrm);

  // layer 1: conv -> relu -> BN stats
  k_gemm128   <<<gemmBlocks, 256, 0, stream>>>(x, W1, bufA, nullptr, nullptr, 0);
  k_agg_init  <<<initBlocks, 256, 0, stream>>>(bufA, deg, b1, bufB);
  k_scatter   <<<ewBlocks, 256, 0, stream>>>(ei, nrm, bufA, bufB);
  k_zero_stats<<<1, 256, 0, stream>>>(stats);
  k_relu_stats<<<1024, 128, 0, stream>>>(bufB, stats);
  k_bn_finalize<<<1, 128, 0, stream>>>(stats, g1, be1, scale, shift);

  // layer 2: conv (BN1 fused into A load) -> relu -> BN stats
  k_gemm128   <<<gemmBlocks, 256, 0, stream>>>(bufB, W2, bufA, scale, shift, 1);
  k_agg_init  <<<initBlocks, 256, 0, stream>>>(bufA, deg, b2, bufB);
  k_scatter   <<<ewBlocks, 256, 0, stream>>>(ei, nrm, bufA, bufB);
  k_zero_stats<<<1, 256, 0, stream>>>(stats);
  k_relu_stats<<<1024, 128, 0, stream>>>(bufB, stats);
  k_bn_finalize<<<1, 128, 0, stream>>>(stats, g2, be2, scale, shift);

  // classifier head (BN2 fused into A load)
  k_gemm_out<<<gemmBlocks, 256, 0, stream>>>(bufB, Wc, bc, scale, shift, out);
}